// LoRA_MoElayer_9766755631796
// MI455X (gfx1250) — compile-verified
//
#include <hip/hip_runtime.h>
#include <hip/hip_bf16.h>
#include <math.h>

typedef __attribute__((ext_vector_type(2))) float v2f;
typedef __attribute__((ext_vector_type(8))) float v8f;

#define E_     7
#define C_     1024
#define DMAX_  128
#define T_     16384            // 8 * 2048 tokens
#define KC_    256              // K-chunk columns (4 chunks of 256 cover C=1024)
#define NCH_   (C_ / KC_)       // 4
#define XS_    260              // LDS row stride (floats); 260 % 64 == 4 -> conflict-free wmma B reads
#define HS_    20               // LDS row stride (floats) for H^T (128 x 16)
#define MAXTILES_ (T_/16 + E_)  // 1031

// ---------------------------------------------------------------- init ----
__global__ void init_counts_kernel(int* __restrict__ counts) {
    if (threadIdx.x < E_) counts[threadIdx.x] = 0;
}

// ---------------------------------------------------------------- gate ----
// One wave32 per token: logits = x_row @ w_gate, nan_to_num, argmax (first max),
// then compact token id into its expert's list.
__global__ __launch_bounds__(256)
void gate_kernel(const float* __restrict__ x, const float* __restrict__ wg,
                 int* __restrict__ counts, int* __restrict__ tlist) {
    int wave = (blockIdx.x * blockDim.x + threadIdx.x) >> 5;
    int lane = threadIdx.x & 31;
    if (wave >= T_) return;
    const float* xr = x + (size_t)wave * C_;

    float acc[E_];
#pragma unroll
    for (int e = 0; e < E_; ++e) acc[e] = 0.f;

#pragma unroll
    for (int i = 0; i < 8; ++i) {
        int c = i * 128 + lane * 4;
        float4 v = *(const float4*)(xr + c);
        const float* w0 = wg + (size_t)c * E_;
#pragma unroll
        for (int e = 0; e < E_; ++e)
            acc[e] += v.x * w0[e] + v.y * w0[E_ + e] + v.z * w0[2 * E_ + e] + v.w * w0[3 * E_ + e];
    }
#pragma unroll
    for (int e = 0; e < E_; ++e) {
#pragma unroll
        for (int off = 16; off >= 1; off >>= 1)
            acc[e] += __shfl_xor(acc[e], off, 32);
    }
    if (lane == 0) {
        int best = 0; float bv = -INFINITY;
#pragma unroll
        for (int e = 0; e < E_; ++e) {
            float v = acc[e];
            if (isnan(v))      v = 0.f;
            else if (isinf(v)) v = (v > 0.f) ? 1e4f : -1e4f;
            if (v > bv) { bv = v; best = e; }   // strict '>' => first index on ties, like top_k
        }
        int slot = atomicAdd(&counts[best], 1);
        tlist[best * T_ + slot] = wave;
    }
}

// ---------------------------------------------------------------- lora ----
// One block = one (expert, 16-token tile). 8 waves.
// GEMM1: H^T(128x16) = A_e(128x1024) @ X^T(1024x16)   (wave w owns rows w*16..w*16+15)
//   X staged via double-buffered async global->LDS copies, pipelined with WMMA.
//   Gather addresses are register-cached once; each chunk issue is addr+imm only.
// GEMM2: OUT^T(1024x16) = Bw_e(1024x128) @ H^T(128x16) (wave w owns 8 c-tiles)
__global__ __launch_bounds__(256)
void lora_kernel(const float* __restrict__ x, const float* __restrict__ A,
                 const float* __restrict__ Bw, const int* __restrict__ counts,
                 const int* __restrict__ tlist, float* __restrict__ y) {
    __shared__ float xbuf[2][16 * XS_];   // 33,280 B double-buffered X chunk
    __shared__ float hts[DMAX_ * HS_];    // 10,240 B H^T
    __shared__ int   toks[16];

    // Map blockIdx.x -> (expert, tile)
    int b = blockIdx.x;
    int e = -1, tile = 0, cnt = 0, accum = 0;
#pragma unroll
    for (int i = 0; i < E_; ++i) {
        int c  = counts[i];
        int nt = (c + 15) >> 4;
        if (e < 0 && b < accum + nt) { e = i; tile = b - accum; cnt = c; }
        accum += nt;
    }
    if (e < 0) return;
    int rows_valid = min(16, cnt - tile * 16);

    if (threadIdx.x < 16) {
        int r  = threadIdx.x;
        int rr = min(r, rows_valid - 1);           // clamp: duplicate a valid row for padding
        toks[r] = tlist[e * T_ + tile * 16 + rr];
    }
    __syncthreads();

    int tid  = threadIdx.x;
    int wave = tid >> 5;
    int lane = tid & 31;
    int lm   = lane & 15;      // fragment M / N index
    int lg   = lane >> 4;      // lane group (0: lanes 0-15, 1: lanes 16-31)

    const float* Ae = A  + (size_t)e * DMAX_ * C_;
    const float* Be = Bw + (size_t)e * C_ * DMAX_;

    // Register-cache this thread's 4 gather addresses (global) and 4 LDS dest
    // offsets ONCE; chunk/buffer selection is pure register arithmetic after this.
    const float* gp[4];
    unsigned int lo[4];
    {
        unsigned int lb0 = (unsigned int)(size_t)&xbuf[0][0];  // low 32 bits = LDS offset
#pragma unroll
        for (int i = 0; i < 4; ++i) {
            int idx = tid + i * 256;            // 0..1023 float4 slots (16 rows x 64 float4)
            int r   = idx >> 6;
            int cq  = idx & 63;
            gp[i] = x + (size_t)toks[r] * C_ + cq * 4;
            lo[i] = lb0 + (unsigned int)(r * XS_ + cq * 4) * 4u;
        }
    }
    const unsigned int BUFB = (unsigned int)(16 * XS_ * 4);    // bytes between buffers

    v8f hacc = {};   // H^T accumulator for this wave's m-tile, lives across all K chunks

    // Prologue: async-stage chunk 0 into buffer 0
#pragma unroll
    for (int i = 0; i < 4; ++i) {
        asm volatile("global_load_async_to_lds_b128 %0, %1, off"
                     :: "v"(lo[i]), "v"(gp[i]) : "memory");
    }
    asm volatile("s_wait_asynccnt 0x0" ::: "memory");
    __syncthreads();

    const float* ArowBase = Ae + (size_t)(wave * 16 + lm) * C_ + 2 * lg;
#pragma unroll
    for (int ch = 0; ch < NCH_; ++ch) {
        if (ch + 1 < NCH_) {               // prefetch next chunk into the other buffer
            unsigned int bsel = (unsigned int)((ch + 1) & 1) * BUFB;
#pragma unroll
            for (int i = 0; i < 4; ++i) {
                const float* g = gp[i] + (ch + 1) * KC_;
                unsigned int la = lo[i] + bsel;
                asm volatile("global_load_async_to_lds_b128 %0, %1, off"
                             :: "v"(la), "v"(g) : "memory");
            }
        }

        // GEMM1 partial: 64 k-steps of 16x16x4 f32 WMMA on current buffer
        const float* Arow = ArowBase + ch * KC_;
        const float* Xrow = &xbuf[ch & 1][lm * XS_ + 2 * lg];
#pragma unroll 4
        for (int ks = 0; ks < KC_ / 4; ++ks) {
            v2f afrag = *(const v2f*)(Arow + ks * 4);   // A_e[m][k], k = 4ks + 2*lg + j
            v2f bfrag = *(const v2f*)(Xrow + ks * 4);   // X^T[k][token lm]
            hacc = __builtin_amdgcn_wmma_f32_16x16x4_f32(
                       false, afrag, false, bfrag, (short)0, hacc, false, false);
        }
        asm volatile("s_wait_asynccnt 0x0" ::: "memory");  // prefetch landed
        __syncthreads();                                    // all waves done reading buffer ch&1
    }

    // Park H^T into LDS: vgpr r holds H^T[wave*16 + r + 8*lg][lm]
#pragma unroll
    for (int r = 0; r < 8; ++r)
        hts[(wave * 16 + r + 8 * lg) * HS_ + lm] = hacc[r];
    __syncthreads();

    // GEMM2: this wave computes 8 c-tiles of OUT^T
    int  tok   = toks[lm];
    bool valid = (lm < rows_valid);
    for (int it = 0; it < 8; ++it) {
        int c0 = (wave * 8 + it) * 16;
        v8f oacc = {};
        const float* Brow = Be + (size_t)(c0 + lm) * DMAX_ + 2 * lg;
#pragma unroll 4
        for (int ks = 0; ks < 32; ++ks) {
            int k0 = ks * 4;
            v2f afrag = *(const v2f*)(Brow + k0);       // Bw_e[c][d]
            v2f bfrag;
            bfrag[0] = hts[(k0 + 2 * lg) * HS_ + lm];   // H^T[d][token lm]
            bfrag[1] = hts[(k0 + 2 * lg + 1) * HS_ + lm];
            oacc = __builtin_amdgcn_wmma_f32_16x16x4_f32(
                       false, afrag, false, bfrag, (short)0, oacc, false, false);
        }
        // Epilogue: exp -> (==0 ? eps) -> log -> nan_to_num -> clip   (matches reference)
#pragma unroll
        for (int r = 0; r < 8; ++r) {
            float v  = oacc[r];
            float cb = expf(v);
            if (cb == 0.f) cb = 2.220446049250313e-16f;   // np.finfo(float64).eps
            float yv = logf(cb);
            if (isnan(yv))      yv = 0.f;
            else if (isinf(yv)) yv = (yv > 0.f) ? 1e4f : -1e4f;
            yv = fminf(fmaxf(yv, -1000.f), 1000.f);
            oacc[r] = yv;
        }
        if (valid) {
            float* yp = y + (size_t)tok * C_ + c0 + 8 * lg;   // channels c0+8*lg .. +7, contiguous
            float4 lo4 = make_float4(oacc[0], oacc[1], oacc[2], oacc[3]);
            float4 hi4 = make_float4(oacc[4], oacc[5], oacc[6], oacc[7]);
            *(float4*)yp       = lo4;
            *((float4*)yp + 1) = hi4;
        }
    }
}

// ---------------------------------------------------------------- loss ----
// importance == load == per-expert counts (gates are exactly 1.0, K=1):
// loss = 2 * cv_squared(counts)
__global__ void loss_kernel(const int* __restrict__ counts, float* __restrict__ out_loss) {
    float v[E_], m = 0.f, s2 = 0.f;
#pragma unroll
    for (int i = 0; i < E_; ++i) { v[i] = (float)counts[i]; m += v[i]; }
    m /= (float)E_;
#pragma unroll
    for (int i = 0; i < E_; ++i) s2 += (v[i] - m) * (v[i] - m);
    s2 /= (float)(E_ - 1);          // ddof=1
    float cv = s2 / (m * m + 1e-10f);
    *out_loss = 2.f * cv;
}

// -------------------------------------------------------------- launch ----
extern "C" void kernel_launch(void* const* d_in, const int* in_sizes, int n_in,
                              void* d_out, int out_size, void* d_ws, size_t ws_size,
                              hipStream_t stream) {
    const float* x  = (const float*)d_in[0];   // [8,2048,1024]
    const float* wg = (const float*)d_in[1];   // [1024,7]
    const float* A  = (const float*)d_in[2];   // [7,128,1024]
    const float* Bw = (const float*)d_in[3];   // [7,1024,128]
    float* y = (float*)d_out;                  // [T*C] then 1 loss float

    int* counts = (int*)d_ws;                  // E ints (zeroed each launch)
    int* tlist  = (int*)d_ws + 8;              // E*T ints

    init_counts_kernel<<<1, 32, 0, stream>>>(counts);
    gate_kernel<<<T_ / 8, 256, 0, stream>>>(x, wg, counts, tlist);
    lora_kernel<<<MAXTILES_, 256, 0, stream>>>(x, A, Bw, counts, tlist, y);
    loss_kernel<<<1, 1, 0, stream>>>(counts, y + (size_t)T_ * C_);
}